// YOLOv1Loss_15530601742762
// MI455X (gfx1250) — compile-verified
//
#include <hip/hip_runtime.h>

#define SGRID   14
#define NCCLS   20
#define NCHAN   30          // 5*B + C
#define NBATCH  4096
#define NCELLS  (NBATCH * SGRID * SGRID)   // 802816
#define K1_BLOCKS  784      // 802816 / (784*256) = 4 cells per thread exactly
#define K1_THREADS 256

typedef __attribute__((ext_vector_type(2))) float v2f;
typedef __attribute__((ext_vector_type(8))) float v8f;

// Wave32 sum via V_WMMA_F32_16X16X4_F32.
// A = ones(16x4), B = {v in VGPR0, 0 in VGPR1}  =>  D[m][n] = colsum(B)[n] for every m.
// d[0] therefore holds the 16 column sums, replicated across lanes 0-15 / 16-31.
// Summing the 16 distinct values (4 xor-shuffles within each half) yields the full
// 32-lane sum (exact: every lane's v lands in exactly one (k,n) slot of B).
__device__ __forceinline__ float wave_sum_wmma(float v) {
    v2f a; a[0] = 1.0f; a[1] = 1.0f;
    v2f b; b[0] = v;    b[1] = 0.0f;
    v8f c = {};
    v8f d = __builtin_amdgcn_wmma_f32_16x16x4_f32(
        /*neg_a=*/false, a, /*neg_b=*/false, b,
        /*c_mod=*/(short)0, c, /*reuse_a=*/false, /*reuse_b=*/false);
    float s = d[0];
    s += __shfl_xor(s, 1, 32);
    s += __shfl_xor(s, 2, 32);
    s += __shfl_xor(s, 4, 32);
    s += __shfl_xor(s, 8, 32);
    return s;   // all lanes hold the wave total
}

// Deterministic block reduction (256 threads = 8 wave32). Result valid in wave 0.
__device__ __forceinline__ float block_sum(float v) {
    __shared__ float smem[8];
    const int lane = threadIdx.x & 31;
    const int wid  = threadIdx.x >> 5;
    float s = wave_sum_wmma(v);
    if (lane == 0) smem[wid] = s;
    __syncthreads();
    float r = 0.0f;
    if (wid == 0) {
        float x = (lane < (int)(blockDim.x >> 5)) ? smem[lane] : 0.0f;
        r = wave_sum_wmma(x);
    }
    return r;
}

__global__ void __launch_bounds__(K1_THREADS)
yolo_loss_partial(const float* __restrict__ pred,
                  const float* __restrict__ target,
                  float* __restrict__ partial) {
    const float invS = 1.0f / (float)SGRID;
    float acc = 0.0f;

    for (int cell = blockIdx.x * blockDim.x + threadIdx.x; cell < NCELLS;
         cell += gridDim.x * blockDim.x) {
        const v2f* P2 = reinterpret_cast<const v2f*>(pred   + (size_t)cell * NCHAN);
        const v2f* T2 = reinterpret_cast<const v2f*>(target + (size_t)cell * NCHAN);

        float p[NCHAN], t[NCHAN];
        #pragma unroll
        for (int i = 0; i < NCHAN / 2; ++i) {      // 15 x b64, non-temporal (streamed once)
            v2f q = __builtin_nontemporal_load(&P2[i]);
            p[2 * i] = q[0]; p[2 * i + 1] = q[1];
        }
        #pragma unroll
        for (int i = 0; i < NCHAN / 2; ++i) {
            v2f q = __builtin_nontemporal_load(&T2[i]);
            t[2 * i] = q[0]; t[2 * i + 1] = q[1];
        }

        // target box (channels 0..4; channels 5..9 are an identical tile copy)
        float tcx = t[0] * invS, tcy = t[1] * invS;
        float thw = 0.5f * t[2], thh = 0.5f * t[3];
        float tx0 = tcx - thw, ty0 = tcy - thh, tx1 = tcx + thw, ty1 = tcy + thh;
        float area_t = (tx1 - tx0) * (ty1 - ty0);

        float iou[2];
        #pragma unroll
        for (int k = 0; k < 2; ++k) {
            float px = p[5 * k + 0], py = p[5 * k + 1];
            float pw = p[5 * k + 2], ph = p[5 * k + 3];
            float cx = px * invS, cy = py * invS, hw = 0.5f * pw, hh = 0.5f * ph;
            float x0 = cx - hw, y0 = cy - hh, x1 = cx + hw, y1 = cy + hh;
            float ltx = fmaxf(x0, tx0), lty = fmaxf(y0, ty0);
            float rbx = fminf(x1, tx1), rby = fminf(y1, ty1);
            float w = fmaxf(rbx - ltx, 0.0f), h = fmaxf(rby - lty, 0.0f);
            float inter  = w * h;
            float area_p = (x1 - x0) * (y1 - y0);
            iou[k] = inter / (area_p + area_t - inter);   // no-obj: 0/area_p = 0, no NaN
        }
        const int   kresp   = (iou[1] > iou[0]) ? 1 : 0;  // argmax, first-on-tie
        const float max_iou = fmaxf(iou[0], iou[1]);
        const bool  obj     = t[4] > 0.0f;

        float cell_loss;
        if (obj) {
            const int b5 = 5 * kresp;
            float dx = p[b5 + 0] - t[0];
            float dy = p[b5 + 1] - t[1];
            float sw = sqrtf(p[b5 + 2]) - sqrtf(t[2]);
            float sh = sqrtf(p[b5 + 3]) - sqrtf(t[3]);
            float dc = p[b5 + 4] - max_iou;
            float lcls = 0.0f;
            #pragma unroll
            for (int c = 0; c < NCCLS; ++c) {
                float d = p[10 + c] - t[10 + c];
                lcls = fmaf(d, d, lcls);
            }
            cell_loss = 5.0f * (dx * dx + dy * dy + sw * sw + sh * sh) + dc * dc + lcls;
        } else {
            // noobj: target conf is 0 -> LAMBDA_NOOBJ * (conf^2 for both boxes)
            cell_loss = 0.5f * (p[4] * p[4] + p[9] * p[9]);
        }
        acc += cell_loss;
    }

    float bsum = block_sum(acc);
    if (threadIdx.x == 0) partial[blockIdx.x] = bsum;
}

__global__ void __launch_bounds__(256)
yolo_loss_final(const float* __restrict__ partial, float* __restrict__ out, int n) {
    float acc = 0.0f;
    for (int i = threadIdx.x; i < n; i += blockDim.x) acc += partial[i];   // fixed order
    float s = block_sum(acc);
    if (threadIdx.x == 0) out[0] = s * (1.0f / (float)NBATCH);
}

extern "C" void kernel_launch(void* const* d_in, const int* in_sizes, int n_in,
                              void* d_out, int out_size, void* d_ws, size_t ws_size,
                              hipStream_t stream) {
    const float* pred    = (const float*)d_in[0];
    const float* target  = (const float*)d_in[1];
    float*       partial = (float*)d_ws;          // K1_BLOCKS floats (3.1 KB)

    yolo_loss_partial<<<K1_BLOCKS, K1_THREADS, 0, stream>>>(pred, target, partial);
    yolo_loss_final<<<1, 256, 0, stream>>>(partial, (float*)d_out, K1_BLOCKS);
}